// TemporalGCN_19902878450282
// MI455X (gfx1250) — compile-verified
//
#include <hip/hip_runtime.h>

// ---------------------------------------------------------------------------
// 2-layer GCN for gfx1250 (MI455X).
//   layer: h = x@W (fp32 WMMA v_wmma_f32_16x16x4_f32, exact),
//          agg = scatter(norm * h[src] -> dst)  [L2-resident, fp32 HW atomics],
//          out = relu(agg + h*dis^2 + b)
// ---------------------------------------------------------------------------

typedef __attribute__((ext_vector_type(2))) float v2f;
typedef __attribute__((ext_vector_type(8))) float v8f;

#define GCN_N 65536
#define GCN_F 64

static __device__ __forceinline__ void atomic_fadd(float* p, float v) {
  __hip_atomic_fetch_add(p, v, __ATOMIC_RELAXED, __HIP_MEMORY_SCOPE_AGENT);
}

// ---------------- utility fills ----------------
__global__ __launch_bounds__(256) void k_zero_u32(unsigned int* __restrict__ p, int n) {
  int t = blockIdx.x * 256 + threadIdx.x;
  if (t < n) p[t] = 0u;
}

__global__ __launch_bounds__(256) void k_zero_f4(float4* __restrict__ p, int n4) {
  int t = blockIdx.x * 256 + threadIdx.x;
  if (t < n4) p[t] = make_float4(0.f, 0.f, 0.f, 0.f);
}

// ---------------- degree / normalization ----------------
__global__ __launch_bounds__(256) void k_deg_count(const long long* __restrict__ dst,
                                                   unsigned int* __restrict__ cnt, int nE) {
  int t = blockIdx.x * 256 + threadIdx.x;
  if (t < nE) atomicAdd(&cnt[(int)dst[t]], 1u);
}

__global__ __launch_bounds__(256) void k_deg_rsqrt(const unsigned int* __restrict__ cnt,
                                                   float* __restrict__ dis, int n) {
  int t = blockIdx.x * 256 + threadIdx.x;
  if (t < n) dis[t] = rsqrtf((float)cnt[t] + 1.0f);
}

// ---------------- fp32 WMMA GEMM: H[N,64] = X[N,64] @ W[64,64] ----------------
// One wave computes a 16-row x 64-col strip using 4 accumulators of
// v_wmma_f32_16x16x4_f32, iterating K=64 in steps of 4.
// A (16x4 f32): lanes 0-15 hold {K=0,K=1} of row M=lane; lanes 16-31 hold {K=2,K=3}.
// B (4x16 f32): lanes 0-15 hold {K=0,K=1} of col N=lane; lanes 16-31 hold {K=2,K=3}.
// C/D (16x16 f32): lane<16 -> VGPR r = D[M=r][N=lane]; lane>=16 -> D[M=r+8][N=lane-16].
__global__ __launch_bounds__(256) void k_gemm64_wmma(const float* __restrict__ X,
                                                     const float* __restrict__ W,
                                                     float* __restrict__ H, int nRows) {
  int waveId  = (blockIdx.x * 256 + threadIdx.x) >> 5;
  int lane    = threadIdx.x & 31;
  int rowBase = waveId * 16;
  if (rowBase >= nRows) return;

  int halfSel = lane >> 4;   // 0: lanes 0-15, 1: lanes 16-31
  int l16     = lane & 15;

  v8f acc0 = {}, acc1 = {}, acc2 = {}, acc3 = {};

  const float* xrow = X + (size_t)(rowBase + l16) * GCN_F;

#pragma unroll
  for (int k0 = 0; k0 < GCN_F; k0 += 4) {
    int kk = k0 + 2 * halfSel;
    v2f a;
    a.x = xrow[kk + 0];
    a.y = xrow[kk + 1];

    const float* w0 = W + (size_t)(kk + 0) * GCN_F;
    const float* w1 = W + (size_t)(kk + 1) * GCN_F;

    v2f b0; b0.x = w0[l16 +  0]; b0.y = w1[l16 +  0];
    v2f b1; b1.x = w0[l16 + 16]; b1.y = w1[l16 + 16];
    v2f b2; b2.x = w0[l16 + 32]; b2.y = w1[l16 + 32];
    v2f b3; b3.x = w0[l16 + 48]; b3.y = w1[l16 + 48];

    acc0 = __builtin_amdgcn_wmma_f32_16x16x4_f32(false, a, false, b0, (short)0, acc0, false, false);
    acc1 = __builtin_amdgcn_wmma_f32_16x16x4_f32(false, a, false, b1, (short)0, acc1, false, false);
    acc2 = __builtin_amdgcn_wmma_f32_16x16x4_f32(false, a, false, b2, (short)0, acc2, false, false);
    acc3 = __builtin_amdgcn_wmma_f32_16x16x4_f32(false, a, false, b3, (short)0, acc3, false, false);
  }

#pragma unroll
  for (int r = 0; r < 8; ++r) {
    int m = r + 8 * halfSel;
    float* o = H + (size_t)(rowBase + m) * GCN_F + l16;
    o[0]  = acc0[r];
    o[16] = acc1[r];
    o[32] = acc2[r];
    o[48] = acc3[r];
  }
}

// ---------------- edge scatter: agg[dst] += h[src] * dis[src]*dis[dst] ----------------
// Each wave owns 32 edges: lane l loads edge (eBase+l)'s indices + norm ONCE,
// then the wave iterates the 32 edges, broadcasting (s, d, norm) via shuffles;
// all 32 lanes then move float2 each (32*2 = 64 features) with HW fp32 atomics.
__global__ __launch_bounds__(256) void k_scatter(const float* __restrict__ H,
                                                 const float* __restrict__ dis,
                                                 const long long* __restrict__ src,
                                                 const long long* __restrict__ dst,
                                                 float* __restrict__ agg, int nE) {
  int lane  = threadIdx.x & 31;
  int wave  = (blockIdx.x * 256 + threadIdx.x) >> 5;
  int eBase = wave * 32;
  if (eBase >= nE) return;

  int myS = 0, myD = 0;
  float myNorm = 0.f;
  int e = eBase + lane;
  if (e < nE) {
    myS = (int)src[e];
    myD = (int)dst[e];
    myNorm = dis[myS] * dis[myD];
  }

  int cnt = nE - eBase;
  if (cnt > 32) cnt = 32;

  for (int i = 0; i < cnt; ++i) {
    int   s   = __shfl(myS, i, 32);
    int   d   = __shfl(myD, i, 32);
    float nrm = __shfl(myNorm, i, 32);

    float2 hv = ((const float2*)(H + (size_t)s * GCN_F))[lane];
    float* ap = agg + (size_t)d * GCN_F + lane * 2;
    atomic_fadd(ap + 0, hv.x * nrm);
    atomic_fadd(ap + 1, hv.y * nrm);
  }
}

// ---------------- finalize: out = relu(agg + h*dis^2 + b) ----------------
__global__ __launch_bounds__(256) void k_finalize(const float* __restrict__ agg,
                                                  const float* __restrict__ H,
                                                  const float* __restrict__ dis,
                                                  const float* __restrict__ bias,
                                                  float* __restrict__ out, int n) {
  int t = blockIdx.x * 256 + threadIdx.x;
  if (t >= n * GCN_F) return;
  int i = t >> 6;
  int f = t & 63;
  float di = dis[i];
  float v  = agg[t] + H[t] * di * di + bias[f];
  out[t] = v > 0.f ? v : 0.f;
}

// ---------------------------------------------------------------------------
extern "C" void kernel_launch(void* const* d_in, const int* in_sizes, int n_in,
                              void* d_out, int out_size, void* d_ws, size_t ws_size,
                              hipStream_t stream) {
  const float*     x   = (const float*)d_in[0];
  const long long* ei  = (const long long*)d_in[1];   // int64 edge_index [2, E]
  const float*     W1  = (const float*)d_in[2];
  const float*     b1  = (const float*)d_in[3];
  const float*     W2  = (const float*)d_in[4];
  const float*     b2  = (const float*)d_in[5];

  const int E = in_sizes[1] / 2;
  const long long* srcIdx = ei;        // row 0
  const long long* dstIdx = ei + E;    // row 1

  // workspace carve
  char* ws = (char*)d_ws;
  unsigned int* cnt = (unsigned int*)ws;                       // N * 4B
  float* dis = (float*)(ws + (size_t)GCN_N * 4);               // N * 4B
  float* h   = (float*)(ws + (size_t)GCN_N * 8);               // 16MB
  float* agg = h + (size_t)GCN_N * GCN_F;                      // 16MB
  float* y1  = agg + (size_t)GCN_N * GCN_F;                    // 16MB

  const int nodeBlocks  = GCN_N / 256;                   // 256
  const int edgeBlocks  = (E + 255) / 256;               // 4096
  const int scatWaves   = (E + 31) / 32;                 // one wave per 32 edges
  const int scatBlocks  = (scatWaves + 7) / 8;           // 8 waves per block
  const int aggBlocks   = (GCN_N * GCN_F / 4) / 256;     // zero via float4
  const int gemmBlocks  = GCN_N / 128;                   // 8 waves * 16 rows
  const int finBlocks   = (GCN_N * GCN_F) / 256;

  // --- degree / normalization (shared by both layers) ---
  k_zero_u32 <<<nodeBlocks, 256, 0, stream>>>(cnt, GCN_N);
  k_deg_count<<<edgeBlocks, 256, 0, stream>>>(dstIdx, cnt, E);
  k_deg_rsqrt<<<nodeBlocks, 256, 0, stream>>>(cnt, dis, GCN_N);

  // --- layer 1 ---
  k_gemm64_wmma<<<gemmBlocks, 256, 0, stream>>>(x, W1, h, GCN_N);
  k_zero_f4   <<<aggBlocks, 256, 0, stream>>>((float4*)agg, GCN_N * GCN_F / 4);
  k_scatter   <<<scatBlocks, 256, 0, stream>>>(h, dis, srcIdx, dstIdx, agg, E);
  k_finalize  <<<finBlocks, 256, 0, stream>>>(agg, h, dis, b1, y1, GCN_N);

  // --- layer 2 ---
  k_gemm64_wmma<<<gemmBlocks, 256, 0, stream>>>(y1, W2, h, GCN_N);
  k_zero_f4   <<<aggBlocks, 256, 0, stream>>>((float4*)agg, GCN_N * GCN_F / 4);
  k_scatter   <<<scatBlocks, 256, 0, stream>>>(h, dis, srcIdx, dstIdx, agg, E);
  k_finalize  <<<finBlocks, 256, 0, stream>>>(agg, h, dis, b2, (float*)d_out, GCN_N);
}